// Feature_Embedding_51496657879142
// MI455X (gfx1250) — compile-verified
//
#include <hip/hip_runtime.h>
#include <stdint.h>

#define FIELDS   16
#define LATENT   64
#define NFEAT    100000
#define NPAIR    120            // 16*15/2
#define ROW_OUT  (NPAIR*LATENT + FIELDS*LATENT)   // 8704 floats per batch row
#define THREADS  256

typedef float f4 __attribute__((ext_vector_type(4)));

__global__ __launch_bounds__(THREADS)
void ffm_interact_kernel(const int* __restrict__ x,
                         const float* __restrict__ W,
                         float* __restrict__ out)
{
    // E[vec][d] with vec = i*16 + j  holds  W[i, x[b,j], d]   (64 KB)
    __shared__ __align__(16) float E[FIELDS * FIELDS * LATENT];

    const int b   = blockIdx.x;
    const int tid = threadIdx.x;

    // ---------------- Gather phase: async global -> LDS, b128 granularity ----
    // 256 vectors * 64 floats = 4096 float4 transfers; 16 lanes serve one
    // 256-byte embedding row, 16 rows per step, 16 steps.
    const int lane16 = tid & 15;   // which float4 of the row
    const int vslot  = tid >> 4;   // which row this step

    #pragma unroll
    for (int s = 0; s < 16; ++s) {
        const int vec = s * 16 + vslot;      // 0..255
        const int i   = vec >> 4;            // latent-table (field) index
        const int j   = vec & 15;            // feature-slot index
        const int idx = x[(size_t)b * FIELDS + j];           // L2-cached broadcast
        const float* src = W + ((size_t)i * NFEAT + (size_t)idx) * LATENT
                             + (size_t)lane16 * 4;
        const uint32_t lds_addr =
            (uint32_t)(uintptr_t)&E[(vec * LATENT) + lane16 * 4];
        const unsigned long long gaddr = (unsigned long long)(uintptr_t)src;
        // CDNA5 async load: memory -> LDS directly, tracked by ASYNCcnt
        asm volatile("global_load_async_to_lds_b128 %0, %1, off"
                     :: "v"(lds_addr), "v"(gaddr)
                     : "memory");
    }

    asm volatile("s_wait_asynccnt 0" ::: "memory");
    __syncthreads();

    // ---------------- Compute + streaming store phase ------------------------
    // 8704 floats = 2176 float4 chunks per row; coalesced, non-temporal stores.
    float* orow = out + (size_t)b * ROW_OUT;

    #pragma unroll 2
    for (int c = tid; c < ROW_OUT / 4; c += THREADS) {
        const int d4 = c & 15;    // float4 within a 64-float vector
        const int p  = c >> 4;    // 0..135 : 120 pairs then 16 diagonals
        f4 r;
        if (p < NPAIR) {
            // decode upper-triangular pair p -> (i, j), i < j (row-major triu)
            int i = 0, pp = p;
            while (pp >= (FIELDS - 1 - i)) { pp -= (FIELDS - 1 - i); ++i; }
            const int j = i + 1 + pp;
            // pair = W[j, x[b,i]] * W[i, x[b,j]] = E[j*16+i] * E[i*16+j]
            const f4 va = *(const f4*)&E[((j * FIELDS + i) * LATENT) + d4 * 4];
            const f4 vb = *(const f4*)&E[((i * FIELDS + j) * LATENT) + d4 * 4];
            r = va * vb;
        } else {
            const int k = p - NPAIR;          // diagonal: W[k, x[b,k]]
            r = *(const f4*)&E[((k * FIELDS + k) * LATENT) + d4 * 4];
        }
        __builtin_nontemporal_store(r, (f4*)(orow + (size_t)c * 4));
    }
}

extern "C" void kernel_launch(void* const* d_in, const int* in_sizes, int n_in,
                              void* d_out, int out_size, void* d_ws, size_t ws_size,
                              hipStream_t stream)
{
    const int*   x = (const int*)d_in[0];      // (4096, 16) int32
    const float* W = (const float*)d_in[1];    // (16, 100000, 64) f32
    float*     out = (float*)d_out;            // (4096, 8704) f32

    const int batch = in_sizes[0] / FIELDS;    // 4096
    ffm_interact_kernel<<<dim3(batch), dim3(THREADS), 0, stream>>>(x, W, out);
}